// GAT_conv_70506183131737
// MI455X (gfx1250) — compile-verified
//
#include <hip/hip_runtime.h>
#include <hip/hip_bf16.h>
#include <math.h>

// Problem constants (fixed by the reference)
#define NNODES 8192
#define F_IN   256
#define UNITS  64
#define HEADS  4
#define HU     (HEADS * UNITS)   // 256 = concat feature width
#define ECAP   1024              // edge cap per row (mean deg ~34; huge safety margin)

typedef __attribute__((ext_vector_type(2))) float v2f;
typedef __attribute__((ext_vector_type(4))) float v4f;
typedef __attribute__((ext_vector_type(8))) float v8f;

// ---------------------------------------------------------------------------
// Kernel 1: feats[n][h*64+u] = sum_f X[n][f] * W[h][f][u]
// One wave per 16x16 output tile, fp32 WMMA, K-loop of 64 x 16x16x4 steps.
// Grid: 1024 blocks x 256 threads (8 waves) = 8192 tiles (512 Mtiles x 16 Ntiles).
// ---------------------------------------------------------------------------
__global__ __launch_bounds__(256) void gat_proj_wmma(
    const float* __restrict__ X,     // [N][F_IN]
    const float* __restrict__ W,     // [H][F_IN][UNITS]
    float* __restrict__ feats)       // [N][HU]
{
    const int wave = threadIdx.x >> 5;
    const int lane = threadIdx.x & 31;
    const int tile = blockIdx.x * 8 + wave;        // 0..8191
    const int mt   = tile >> 4;                    // row tile (0..511)
    const int nt   = tile & 15;                    // col tile (0..15)
    const int row0 = mt << 4;
    const int col0 = nt << 4;
    const int h    = col0 >> 6;                    // head of this col tile
    const int u0   = col0 & 63;

    const int m  = lane & 15;                      // M (A) / N (B) index
    const int kp = (lane >> 4) << 1;               // K pair base: 0 or 2

    const float* __restrict__ Wh = W + h * (F_IN * UNITS);
    const float* __restrict__ Xr = X + (size_t)(row0 + m) * F_IN;

    v8f acc = {};
    #pragma unroll 4
    for (int k0 = 0; k0 < F_IN; k0 += 4) {
        // A fragment: lane (l): M = l&15, K = 2*(l>>4) + v  -> contiguous pair (b64)
        v2f a = *(const v2f*)(Xr + k0 + kp);
        // B fragment: lane (l): N = l&15, K = 2*(l>>4) + v
        v2f b;
        b.x = Wh[(k0 + kp)     * UNITS + u0 + m];
        b.y = Wh[(k0 + kp + 1) * UNITS + u0 + m];
        acc = __builtin_amdgcn_wmma_f32_16x16x4_f32(
            /*neg_a=*/false, a, /*neg_b=*/false, b,
            /*c_mod=*/(short)0, acc, /*reuse_a=*/false, /*reuse_b=*/false);
    }

    // C/D layout: VGPR v: lanes 0-15 -> M=v, lanes 16-31 -> M=v+8; N = lane&15
    const int prow = row0 + ((lane >> 4) << 3);
    const int pcol = col0 + m;
    #pragma unroll
    for (int v = 0; v < 8; ++v)
        feats[(size_t)(prow + v) * HU + pcol] = acc[v];
}

// ---------------------------------------------------------------------------
// Kernel 2: a_self[h][n] = feats[n][h*64:..] . attn_self[h]; same for neigh.
// One wave per node; wave32 shfl_xor tree reduction.
// ---------------------------------------------------------------------------
__global__ __launch_bounds__(256) void gat_attn_coef(
    const float* __restrict__ feats,   // [N][HU]
    const float* __restrict__ asv,     // [H][UNITS]
    const float* __restrict__ anv,     // [H][UNITS]
    float* __restrict__ a_self,        // [H][N]
    float* __restrict__ a_neigh)       // [H][N]
{
    const int wave = threadIdx.x >> 5;
    const int lane = threadIdx.x & 31;
    const int n = blockIdx.x * 8 + wave;
    const float* __restrict__ fr = feats + (size_t)n * HU;

    #pragma unroll
    for (int h = 0; h < HEADS; ++h) {
        const float x0 = fr[h * UNITS + lane];
        const float x1 = fr[h * UNITS + 32 + lane];
        float s = x0 * asv[h * UNITS + lane] + x1 * asv[h * UNITS + 32 + lane];
        float t = x0 * anv[h * UNITS + lane] + x1 * anv[h * UNITS + 32 + lane];
        #pragma unroll
        for (int off = 16; off > 0; off >>= 1) {   // wave32 reduction
            s += __shfl_xor(s, off, 32);
            t += __shfl_xor(t, off, 32);
        }
        if (lane == 0) {
            a_self [h * NNODES + n] = s;
            a_neigh[h * NNODES + n] = t;
        }
    }
}

// ---------------------------------------------------------------------------
// Kernel 3: one block per row i. Stream A-row NON-TEMPORALLY (268 MB > 192 MB
// L2; must not evict the L2-resident feats/a_neigh), compact edges
// deterministically, masked softmax over edges, aggregate feats, ELU,
// non-temporal store of out (write-once, never re-read).
// ---------------------------------------------------------------------------
__global__ __launch_bounds__(256) void gat_aggregate(
    const float* __restrict__ A,        // [N][N] (0/1 floats)
    const float* __restrict__ feats,    // [N][HU]
    const float* __restrict__ a_self,   // [H][N]
    const float* __restrict__ a_neigh,  // [H][N]
    float* __restrict__ out)            // [N][HU]
{
    __shared__ int   s_idx[ECAP];            // compacted edge column indices
    __shared__ float s_p[ECAP][HEADS];       // per-edge scores -> exp weights
    __shared__ float s_red[256][HEADS];      // reduction scratch
    __shared__ int   s_woff[9];              // per-wave exclusive offsets + total
    __shared__ float s_asi[HEADS];           // a_self for this row
    __shared__ float s_m[HEADS];             // per-head max
    __shared__ float s_sum[HEADS];           // per-head softmax denominator

    const int i    = blockIdx.x;
    const int t    = threadIdx.x;
    const int lane = t & 31;
    const int wave = t >> 5;

    if (t < HEADS) s_asi[t] = a_self[t * NNODES + i];

    // ---- Phase 1: stream A row (coalesced b128, TH=NT), per-thread bitmask --
    const v4f* __restrict__ Arow4 = (const v4f*)(A + (size_t)i * NNODES);
    unsigned mask = 0;
    #pragma unroll
    for (int k = 0; k < 8; ++k) {
        v4f a = __builtin_nontemporal_load(&Arow4[k * 256 + t]); // elems t*4+k*1024..+3
        mask |= (unsigned)(a.x != 0.0f) << (4 * k + 0);
        mask |= (unsigned)(a.y != 0.0f) << (4 * k + 1);
        mask |= (unsigned)(a.z != 0.0f) << (4 * k + 2);
        mask |= (unsigned)(a.w != 0.0f) << (4 * k + 3);
    }
    const int cnt = __popc(mask);

    // deterministic hierarchical exclusive scan: shfl scan in wave, serial over 8
    int inc = cnt;
    #pragma unroll
    for (int d = 1; d < 32; d <<= 1) {
        int v = __shfl_up(inc, d, 32);
        if (lane >= d) inc += v;
    }
    if (lane == 31) s_woff[wave] = inc;      // wave total
    __syncthreads();
    if (t == 0) {
        int run = 0;
        #pragma unroll
        for (int w = 0; w < 8; ++w) { int c = s_woff[w]; s_woff[w] = run; run += c; }
        s_woff[8] = run;
    }
    __syncthreads();

    const int E = min(s_woff[8], ECAP);
    int pos = s_woff[wave] + (inc - cnt);    // this thread's exclusive offset
    unsigned mm = mask;
    while (mm) {
        const int b = __ffs(mm) - 1;
        mm &= mm - 1;
        const int j = ((b >> 2) << 10) + (t << 2) + (b & 3);
        if (pos < ECAP) s_idx[pos] = j;
        ++pos;
    }
    __syncthreads();

    // ---- Phase 2a: leaky-relu scores + per-head max ----
    float mloc[HEADS] = { -INFINITY, -INFINITY, -INFINITY, -INFINITY };
    for (int e = t; e < E; e += 256) {
        const int j = s_idx[e];
        #pragma unroll
        for (int h = 0; h < HEADS; ++h) {
            float s = s_asi[h] + a_neigh[h * NNODES + j];
            s = (s > 0.0f) ? s : 0.2f * s;   // leaky_relu(0.2)
            s_p[e][h] = s;
            mloc[h] = fmaxf(mloc[h], s);
        }
    }
    #pragma unroll
    for (int h = 0; h < HEADS; ++h) s_red[t][h] = mloc[h];
    __syncthreads();
    for (int st = 128; st > 0; st >>= 1) {
        if (t < st) {
            #pragma unroll
            for (int h = 0; h < HEADS; ++h)
                s_red[t][h] = fmaxf(s_red[t][h], s_red[t + st][h]);
        }
        __syncthreads();
    }
    if (t < HEADS) s_m[t] = s_red[0][t];
    __syncthreads();

    // ---- Phase 2b: exp(score - max) + per-head denominator ----
    float sloc[HEADS] = { 0.f, 0.f, 0.f, 0.f };
    for (int e = t; e < E; e += 256) {
        #pragma unroll
        for (int h = 0; h < HEADS; ++h) {
            const float p = __expf(s_p[e][h] - s_m[h]);
            s_p[e][h] = p;
            sloc[h] += p;
        }
    }
    #pragma unroll
    for (int h = 0; h < HEADS; ++h) s_red[t][h] = sloc[h];
    __syncthreads();
    for (int st = 128; st > 0; st >>= 1) {
        if (t < st) {
            #pragma unroll
            for (int h = 0; h < HEADS; ++h)
                s_red[t][h] += s_red[t + st][h];
        }
        __syncthreads();
    }
    if (t < HEADS) s_sum[t] = s_red[0][t];
    __syncthreads();

    // ---- Phase 2c: weighted aggregation, normalize, ELU ----
    const int h = t >> 6;                    // t = h*64 + u (HU == 256)
    float acc = 0.0f;
    for (int e = 0; e < E; ++e) {
        if (e + 4 < E)                       // -> global_prefetch_b8
            __builtin_prefetch(&feats[(size_t)s_idx[e + 4] * HU + t], 0, 0);
        acc += s_p[e][h] * feats[(size_t)s_idx[e] * HU + t];
    }
    float r = acc / s_sum[h];
    r = (r > 0.0f) ? r : expm1f(r);          // ELU(alpha=1)
    __builtin_nontemporal_store(r, &out[(size_t)i * HU + t]);
}

// ---------------------------------------------------------------------------
// Host-side launch
// ---------------------------------------------------------------------------
extern "C" void kernel_launch(void* const* d_in, const int* in_sizes, int n_in,
                              void* d_out, int out_size, void* d_ws, size_t ws_size,
                              hipStream_t stream) {
    const float* X   = (const float*)d_in[0];   // [N][F_IN]
    const float* A   = (const float*)d_in[1];   // [N][N]
    const float* W   = (const float*)d_in[2];   // [H][F_IN][UNITS]
    const float* asv = (const float*)d_in[3];   // [H][UNITS]
    const float* anv = (const float*)d_in[4];   // [H][UNITS]
    float* out = (float*)d_out;                 // [N][HU]

    // workspace layout: feats | a_self | a_neigh  (~8.65 MB)
    float* feats   = (float*)d_ws;
    float* a_self  = feats + (size_t)NNODES * HU;
    float* a_neigh = a_self + (size_t)HEADS * NNODES;

    gat_proj_wmma<<<(NNODES / 16) * (HU / 16) / 8, 256, 0, stream>>>(X, W, feats);
    gat_attn_coef<<<NNODES / 8, 256, 0, stream>>>(feats, asv, anv, a_self, a_neigh);
    gat_aggregate<<<NNODES, 256, 0, stream>>>(A, feats, a_self, a_neigh, out);
}